// GraphConvLayer_23347442221695
// MI455X (gfx1250) — compile-verified
//
#include <hip/hip_runtime.h>
#include <math.h>

typedef __attribute__((ext_vector_type(2))) float v2f;
typedef __attribute__((ext_vector_type(8))) float v8f;

#define NB 8
#define NN 256
#define ND 128
#define LN_EPS 1e-5f

__device__ __forceinline__ v8f wmma_f32_k4(v2f a, v2f b, v8f c) {
  // D = A(16x4,f32) * B(4x16,f32) + C(16x16,f32), full fp32 precision
  return __builtin_amdgcn_wmma_f32_16x16x4_f32(false, a, false, b, (short)0, c,
                                               false, false);
}

// ---------------------------------------------------------------------------
// Kernel A: h = X@Wn + bn ; a_i = X@We1[:128] ; a_j = X@We1[128:256]
// X is (B*N, 128) = (2048,128). One wave per 16x16 output tile.
// Tiles: mt in [0,128), nt in [0,24): nt/8 selects projection, nt%8 the n-tile.
// ---------------------------------------------------------------------------
__global__ void proj_gemm(const float* __restrict__ X,
                          const float* __restrict__ Wn,
                          const float* __restrict__ bn,
                          const float* __restrict__ We1,
                          float* __restrict__ h,
                          float* __restrict__ ai,
                          float* __restrict__ aj) {
  int wave = threadIdx.x >> 5;
  int lane = threadIdx.x & 31;
  int tile = blockIdx.x * 4 + wave;       // 768 blocks * 4 waves = 3072 tiles
  int mt = tile / 24;
  int nt = tile % 24;
  int proj = nt >> 3;
  int n0 = (nt & 7) << 4;
  int m0 = mt << 4;
  int half = lane >> 4;
  int col = lane & 15;

  const float* Bmat = (proj == 0) ? Wn : (proj == 1 ? We1 : We1 + 128 * 128);
  float* Out = (proj == 0) ? h : (proj == 1 ? ai : aj);

  const float* Arow = X + (size_t)(m0 + col) * ND + 2 * half;
  const float* Bcol = Bmat + (size_t)(2 * half) * ND + n0 + col;

  v8f acc = {};
#pragma unroll 4
  for (int k = 0; k < ND; k += 4) {
    v2f a = *(const v2f*)(Arow + k);      // A[col][k+2h], A[col][k+2h+1]
    v2f bv;
    bv.x = Bcol[(size_t)k * ND];          // B[k+2h][n0+col]
    bv.y = Bcol[(size_t)(k + 1) * ND];    // B[k+2h+1][n0+col]
    acc = wmma_f32_k4(a, bv, acc);
  }
  float bias = (proj == 0) ? bn[n0 + col] : 0.0f;
#pragma unroll
  for (int r = 0; r < 8; ++r) {
    int row = m0 + r + 8 * half;
    Out[(size_t)row * ND + n0 + col] = acc[r] + bias;
  }
}

// ---------------------------------------------------------------------------
// Kernel B: edge MLP + masked softmax. One 256-thread block per (b,i).
// score(i,j) = relu((a_i+be1) + a_j + rx*W1 + ry*W2 + dist*W3) . We2 + be2
// Per-lane d-slice is fixed -> all six operand slices live in registers.
// ---------------------------------------------------------------------------
__global__ void edge_softmax(const float* __restrict__ ai,
                             const float* __restrict__ aj,
                             const float* __restrict__ rel,   // (B,N,N,2)
                             const int* __restrict__ adj,     // (B,N,N)
                             const float* __restrict__ We1,   // rows 256..258
                             const float* __restrict__ be1,
                             const float* __restrict__ We2,   // (128,1)
                             const float* __restrict__ be2,   // (1,)
                             float* __restrict__ wgt) {       // (B,N,N)
  __shared__ float s_sc[NN];
  __shared__ float s_red[8];
  int bi = blockIdx.x;                    // b*N + i, 0..2047
  int tid = threadIdx.x;
  int wave = tid >> 5, lane = tid & 31;
  int d0 = lane * 4;

  // Loop-invariant per-lane operands: coalesced float4 register loads.
  float4 rai = *(const float4*)(ai + (size_t)bi * ND + d0);
  float4 rb1 = *(const float4*)(be1 + d0);
  float4 rw1 = *(const float4*)(We1 + (size_t)256 * ND + d0);
  float4 rw2 = *(const float4*)(We1 + (size_t)257 * ND + d0);
  float4 rw3 = *(const float4*)(We1 + (size_t)258 * ND + d0);
  float4 rv2 = *(const float4*)(We2 + d0);
  float base[4] = {rai.x + rb1.x, rai.y + rb1.y, rai.z + rb1.z, rai.w + rb1.w};
  float w1[4] = {rw1.x, rw1.y, rw1.z, rw1.w};
  float w2[4] = {rw2.x, rw2.y, rw2.z, rw2.w};
  float w3[4] = {rw3.x, rw3.y, rw3.z, rw3.w};
  float v2[4] = {rv2.x, rv2.y, rv2.z, rv2.w};

  float bias2 = be2[0];
  int b = bi >> 8;
  const float* ajb = aj + (size_t)b * NN * ND;
  const float* relrow = rel + (size_t)bi * NN * 2;
  const int* adjrow = adj + (size_t)bi * NN;

  for (int t = 0; t < 32; ++t) {          // 8 waves * 32 iters = 256 j
    int j = (t << 3) | wave;
    if (t + 1 < 32) {                     // keep L0 warm one row ahead
      __builtin_prefetch(ajb + (size_t)(j + 8) * ND + d0, 0, 1);
    }
    float4 av = *(const float4*)(ajb + (size_t)j * ND + d0);
    float2 r2 = *(const float2*)(relrow + 2 * j);
    float rx = r2.x, ry = r2.y;
    float dist = sqrtf(rx * rx + ry * ry);
    float avv[4] = {av.x, av.y, av.z, av.w};
    float p = 0.f;
#pragma unroll
    for (int u = 0; u < 4; ++u) {
      float hd = base[u] + avv[u] + rx * w1[u] + ry * w2[u] + dist * w3[u];
      hd = fmaxf(hd, 0.f);
      p += hd * v2[u];
    }
#pragma unroll
    for (int m = 16; m >= 1; m >>= 1) p += __shfl_xor(p, m, 32);
    if (lane == 0) {
      float sc = p + bias2;
      s_sc[j] = (adjrow[j] == 0) ? -INFINITY : sc;
    }
  }
  __syncthreads();

  // softmax over 256 scores: wave shuffle reduce + 8-entry LDS combine
  float sc = s_sc[tid];
  float m = sc;
#pragma unroll
  for (int x = 16; x >= 1; x >>= 1) m = fmaxf(m, __shfl_xor(m, x, 32));
  if (lane == 0) s_red[wave] = m;
  __syncthreads();
  float mx = s_red[0];
#pragma unroll
  for (int x = 1; x < 8; ++x) mx = fmaxf(mx, s_red[x]);
  __syncthreads();

  float e = (sc == -INFINITY) ? 0.f : __expf(sc - mx);
  float s = e;
#pragma unroll
  for (int x = 16; x >= 1; x >>= 1) s += __shfl_xor(s, x, 32);
  if (lane == 0) s_red[wave] = s;
  __syncthreads();
  float sum = 0.f;
#pragma unroll
  for (int x = 0; x < 8; ++x) sum += s_red[x];

  float w = (sum > 0.f) ? e / sum : 0.f;  // fully-masked row -> 0 (NaN guard)
  wgt[(size_t)bi * NN + tid] = w;
}

// ---------------------------------------------------------------------------
// Kernel C: messages = w @ h, per batch M=256, K=256, N=128. One wave/tile.
// ---------------------------------------------------------------------------
__global__ void msg_gemm(const float* __restrict__ wgt,   // (B,N,N)
                         const float* __restrict__ h,     // (B,N,128)
                         float* __restrict__ msg) {       // (B,N,128)
  int wave = threadIdx.x >> 5;
  int lane = threadIdx.x & 31;
  int tile = blockIdx.x * 4 + wave;       // 256 blocks * 4 = 1024 tiles
  int b = tile >> 7;                      // 128 tiles per batch
  int t = tile & 127;
  int m0 = (t >> 3) << 4;
  int n0 = (t & 7) << 4;
  int half = lane >> 4, col = lane & 15;

  const float* A = wgt + (size_t)b * NN * NN;
  const float* Bm = h + (size_t)b * NN * ND;
  const float* Arow = A + (size_t)(m0 + col) * NN + 2 * half;
  const float* Bcol = Bm + (size_t)(2 * half) * ND + n0 + col;

  v8f acc = {};
#pragma unroll 4
  for (int k = 0; k < NN; k += 4) {
    v2f a = *(const v2f*)(Arow + k);
    v2f bv;
    bv.x = Bcol[(size_t)k * ND];
    bv.y = Bcol[(size_t)(k + 1) * ND];
    acc = wmma_f32_k4(a, bv, acc);
  }
  float* Om = msg + (size_t)b * NN * ND;
#pragma unroll
  for (int r = 0; r < 8; ++r)
    Om[(size_t)(m0 + r + 8 * half) * ND + n0 + col] = acc[r];
}

// ---------------------------------------------------------------------------
// Kernel D: out = LN(X + relu(msg @ Wa + ba)) * gamma + beta
// 256 threads = 8 waves; each wave one 16x16 n-tile of a shared 16-row block.
// ---------------------------------------------------------------------------
__global__ void out_gemm_ln(const float* __restrict__ msg,
                            const float* __restrict__ Wa,
                            const float* __restrict__ ba,
                            const float* __restrict__ X,
                            const float* __restrict__ gamma,
                            const float* __restrict__ beta,
                            float* __restrict__ out) {
  __shared__ float s_x[16][ND];
  int tid = threadIdx.x;
  int wave = tid >> 5, lane = tid & 31;
  int m0 = blockIdx.x << 4;
  int n0 = wave << 4;
  int half = lane >> 4, col = lane & 15;

  const float* Arow = msg + (size_t)(m0 + col) * ND + 2 * half;
  const float* Bcol = Wa + (size_t)(2 * half) * ND + n0 + col;
  v8f acc = {};
#pragma unroll 4
  for (int k = 0; k < ND; k += 4) {
    v2f a = *(const v2f*)(Arow + k);
    v2f bv;
    bv.x = Bcol[(size_t)k * ND];
    bv.y = Bcol[(size_t)(k + 1) * ND];
    acc = wmma_f32_k4(a, bv, acc);
  }
  float bias = ba[n0 + col];
#pragma unroll
  for (int r = 0; r < 8; ++r) {
    int row = r + 8 * half;
    float v = fmaxf(acc[r] + bias, 0.f) +
              X[(size_t)(m0 + row) * ND + n0 + col];
    s_x[row][n0 + col] = v;
  }
  __syncthreads();

  // LayerNorm: 16 threads per row (within one wave half), 8 elems/thread.
  int row = tid >> 4, sub = tid & 15;
  float vals[8], lsum = 0.f, lsq = 0.f;
#pragma unroll
  for (int u = 0; u < 8; ++u) {
    float v = s_x[row][sub * 8 + u];
    vals[u] = v;
    lsum += v;
    lsq += v * v;
  }
#pragma unroll
  for (int m = 8; m >= 1; m >>= 1) {
    lsum += __shfl_xor(lsum, m, 16);
    lsq += __shfl_xor(lsq, m, 16);
  }
  float mu = lsum * (1.f / 128.f);
  float var = lsq * (1.f / 128.f) - mu * mu;
  float rstd = rsqrtf(var + LN_EPS);
#pragma unroll
  for (int u = 0; u < 8; ++u) {
    int d = sub * 8 + u;
    out[(size_t)(m0 + row) * ND + d] = (vals[u] - mu) * rstd * gamma[d] + beta[d];
  }
}

// ---------------------------------------------------------------------------
extern "C" void kernel_launch(void* const* d_in, const int* in_sizes, int n_in,
                              void* d_out, int out_size, void* d_ws,
                              size_t ws_size, hipStream_t stream) {
  (void)in_sizes; (void)n_in; (void)out_size; (void)ws_size;
  const float* X = (const float*)d_in[0];
  const int* adj = (const int*)d_in[1];
  const float* rel = (const float*)d_in[2];
  const float* Wn = (const float*)d_in[3];
  const float* bn = (const float*)d_in[4];
  const float* We1 = (const float*)d_in[5];
  const float* be1 = (const float*)d_in[6];
  const float* We2 = (const float*)d_in[7];
  const float* be2 = (const float*)d_in[8];
  const float* Wa = (const float*)d_in[9];
  const float* ba = (const float*)d_in[10];
  const float* gamma = (const float*)d_in[11];
  const float* beta = (const float*)d_in[12];

  float* ws = (float*)d_ws;
  float* h = ws;                  // 2048*128
  float* ai = ws + 262144;        // 2048*128
  float* aj = ws + 524288;        // 2048*128
  float* wgt = ws + 786432;       // 8*256*256
  float* msg = ws + 1310720;      // 2048*128

  proj_gemm<<<768, 128, 0, stream>>>(X, Wn, bn, We1, h, ai, aj);
  edge_softmax<<<2048, 256, 0, stream>>>(ai, aj, rel, adj, We1, be1, We2, be2,
                                         wgt);
  msg_gemm<<<256, 128, 0, stream>>>(wgt, h, msg);
  out_gemm_ln<<<128, 256, 0, stream>>>(msg, Wa, ba, X, gamma, beta,
                                       (float*)d_out);
}